// SimpleGraphLayer_23476291240575
// MI455X (gfx1250) — compile-verified
//
#include <hip/hip_runtime.h>

// N=1024 nodes, D=128 features
constexpr int N     = 1024;
constexpr int D     = 128;
constexpr int TWO_D = 256;
constexpr int JCH   = 4;      // j-chunks for the aggregation split
constexpr int JLEN  = N / JCH;

typedef float v2f __attribute__((ext_vector_type(2)));
typedef float v8f __attribute__((ext_vector_type(8)));

__device__ __forceinline__ float gelu_exact(float x) {
  // exact GELU: x * 0.5 * (1 + erf(x / sqrt(2)))
  return 0.5f * x * (1.0f + erff(x * 0.70710678118654752440f));
}

// ---------------------------------------------------------------------------
// Kernel A: SD[n][c] for c in [0,256):
//   c <  128 : S[n,c]   = sum_k X[n,k] * msg_w[c, k]          (src projection)
//   c >= 128 : DpB[n,c] = sum_k X[n,k] * msg_w[c-128, 128+k] + msg_b[c-128]
// One wave per 16x16 output tile, fp32 WMMA 16x16x4, K-loop over 128.
// 64 M-tiles x 16 C-tiles = 1024 tiles; 4 waves/block -> 256 blocks.
// ---------------------------------------------------------------------------
__global__ __launch_bounds__(128)
void proj_kernel(const float* __restrict__ X, const float* __restrict__ msg_w,
                 const float* __restrict__ msg_b, float* __restrict__ SD) {
  const int lane = threadIdx.x & 31;
  const int wave = threadIdx.x >> 5;
  const int tile = blockIdx.x * 4 + wave;     // 0..1023
  const int tm   = tile >> 4;                 // 0..63
  const int tc   = tile & 15;                 // 0..15
  const int half = lane >> 4;                 // 0 or 1
  const int l15  = lane & 15;
  const int m    = tm * 16 + l15;             // A-matrix row for this lane
  const int c    = tc * 16 + l15;             // B/D column for this lane
  // B[k][c] = (c<128) ? msg_w[c*256 + k] : msg_w[(c-128)*256 + 128 + k]
  const float* brow = (c < D) ? (msg_w + c * TWO_D)
                              : (msg_w + (c - D) * TWO_D + D);
  v8f acc = {};
#pragma unroll 4
  for (int k0 = 0; k0 < D; k0 += 4) {
    const int ka = k0 + half * 2;
    v2f a, b;
    a.x = X[m * D + ka];
    a.y = X[m * D + ka + 1];
    b.x = brow[ka];
    b.y = brow[ka + 1];
    acc = __builtin_amdgcn_wmma_f32_16x16x4_f32(false, a, false, b,
                                                (short)0, acc, false, false);
  }
  const float bias = (c >= D) ? msg_b[c - D] : 0.0f;
#pragma unroll
  for (int r = 0; r < 8; ++r) {
    const int mo = tm * 16 + r + half * 8;    // D-matrix row layout
    SD[mo * TWO_D + c] = acc[r] + bias;
  }
}

// ---------------------------------------------------------------------------
// Kernel B (hot): partial aggregation over a j-chunk.
//   AGGP[jc][i][d] = sum_{j in chunk} adj[i,j] * gelu(S[i,d] + DpB[j,d])
// One wave per i-row: lanes cover d via float4; adj[i,j] is wave-uniform
// (scalar load). Grid (128, JCH), 8 waves/block -> 4096 waves total.
// ---------------------------------------------------------------------------
__global__ __launch_bounds__(256)
void agg_kernel(const float* __restrict__ SD, const float* __restrict__ adj,
                float* __restrict__ AGGP) {
  const int lane = threadIdx.x & 31;
  const int wave = threadIdx.x >> 5;          // 0..7
  const int i    = blockIdx.x * 8 + wave;     // node row
  const int jc   = blockIdx.y;                // j-chunk
  const int d0   = lane * 4;

  const float4 sv = *(const float4*)(SD + i * TWO_D + d0);
  const float* __restrict__ arow = adj + (size_t)i * N;

  float4 acc = make_float4(0.f, 0.f, 0.f, 0.f);
  const int j0 = jc * JLEN;
#pragma unroll 4
  for (int j = j0; j < j0 + JLEN; ++j) {
    const float  aij = arow[j];                                   // uniform
    const float4 dv  = *(const float4*)(SD + j * TWO_D + D + d0); // coalesced
    acc.x += aij * gelu_exact(sv.x + dv.x);
    acc.y += aij * gelu_exact(sv.y + dv.y);
    acc.z += aij * gelu_exact(sv.z + dv.z);
    acc.w += aij * gelu_exact(sv.w + dv.w);
  }
  *(float4*)(AGGP + ((size_t)jc * N + i) * D + d0) = acc;
}

// Deterministic reduction of the JCH partials.
__global__ __launch_bounds__(256)
void agg_reduce(const float* __restrict__ AGGP, float* __restrict__ AGG) {
  const int t = blockIdx.x * 256 + threadIdx.x;   // over N*D
  float s = AGGP[t];
#pragma unroll
  for (int p = 1; p < JCH; ++p) s += AGGP[(size_t)p * N * D + t];
  AGG[t] = s;
}

// ---------------------------------------------------------------------------
// Kernel C: new_feats = [X | AGG] @ upd_w^T + upd_b ; h = gelu(nf) + X
// fp32 WMMA, K=256. 64 M-tiles x 8 C-tiles = 512 tiles; 4 waves/block -> 128.
// ---------------------------------------------------------------------------
__global__ __launch_bounds__(128)
void update_kernel(const float* __restrict__ X, const float* __restrict__ AGG,
                   const float* __restrict__ upd_w, const float* __restrict__ upd_b,
                   float* __restrict__ out) {
  const int lane = threadIdx.x & 31;
  const int wave = threadIdx.x >> 5;
  const int tile = blockIdx.x * 4 + wave;     // 0..511
  const int tm   = tile >> 3;                 // 0..63
  const int tc   = tile & 7;                  // 0..7
  const int half = lane >> 4;
  const int l15  = lane & 15;
  const int m    = tm * 16 + l15;
  const int c    = tc * 16 + l15;

  v8f acc = {};
#pragma unroll 2
  for (int k0 = 0; k0 < TWO_D; k0 += 4) {     // each 4-step is wholly X or AGG
    const int ka = k0 + half * 2;
    v2f a, b;
    if (k0 < D) {
      a.x = X[m * D + ka];
      a.y = X[m * D + ka + 1];
    } else {
      a.x = AGG[m * D + (ka - D)];
      a.y = AGG[m * D + (ka - D) + 1];
    }
    b.x = upd_w[c * TWO_D + ka];
    b.y = upd_w[c * TWO_D + ka + 1];
    acc = __builtin_amdgcn_wmma_f32_16x16x4_f32(false, a, false, b,
                                                (short)0, acc, false, false);
  }
  const float bias = upd_b[c];
#pragma unroll
  for (int r = 0; r < 8; ++r) {
    const int mo = tm * 16 + r + half * 8;
    const float nf = acc[r] + bias;
    out[mo * D + c] = gelu_exact(nf) + X[mo * D + c];
  }
}

// ---------------------------------------------------------------------------
// Kernel D: in-place LayerNorm per row (one 128-thread block per row).
// ---------------------------------------------------------------------------
__global__ __launch_bounds__(128)
void ln_kernel(float* __restrict__ out, const float* __restrict__ ln_w,
               const float* __restrict__ ln_b) {
  __shared__ float s1[128];
  __shared__ float s2[128];
  const int t = threadIdx.x;
  const int n = blockIdx.x;
  const float x = out[n * D + t];
  s1[t] = x;
  s2[t] = x * x;
  __syncthreads();
#pragma unroll
  for (int s = 64; s > 0; s >>= 1) {
    if (t < s) { s1[t] += s1[t + s]; s2[t] += s2[t + s]; }
    __syncthreads();
  }
  const float mu  = s1[0] * (1.0f / (float)D);
  const float var = s2[0] * (1.0f / (float)D) - mu * mu;
  const float inv = rsqrtf(var + 1e-5f);
  out[n * D + t] = (x - mu) * inv * ln_w[t] + ln_b[t];
}

// ---------------------------------------------------------------------------
extern "C" void kernel_launch(void* const* d_in, const int* in_sizes, int n_in,
                              void* d_out, int out_size, void* d_ws, size_t ws_size,
                              hipStream_t stream) {
  (void)in_sizes; (void)n_in; (void)out_size; (void)ws_size;
  const float* X     = (const float*)d_in[0];
  const float* adj   = (const float*)d_in[1];
  const float* msg_w = (const float*)d_in[2];
  const float* msg_b = (const float*)d_in[3];
  const float* upd_w = (const float*)d_in[4];
  const float* upd_b = (const float*)d_in[5];
  const float* ln_w  = (const float*)d_in[6];
  const float* ln_b  = (const float*)d_in[7];
  float* out = (float*)d_out;

  float* SD   = (float*)d_ws;            // N*2D   = 262144 floats (1.0 MB)
  float* AGGP = SD + N * TWO_D;          // JCH*N*D = 524288 floats (2.0 MB)
  float* AGG  = AGGP + JCH * N * D;      // N*D    = 131072 floats (0.5 MB)

  proj_kernel  <<<256, 128, 0, stream>>>(X, msg_w, msg_b, SD);
  agg_kernel   <<<dim3(N / 8, JCH), 256, 0, stream>>>(SD, adj, AGGP);
  agg_reduce   <<<(N * D) / 256, 256, 0, stream>>>(AGGP, AGG);
  update_kernel<<<128, 128, 0, stream>>>(X, AGG, upd_w, upd_b, out);
  ln_kernel    <<<N, 128, 0, stream>>>(out, ln_w, ln_b);
}